// SageConvExt_55576876810841
// MI455X (gfx1250) — compile-verified
//
#include <hip/hip_runtime.h>

typedef __attribute__((ext_vector_type(2))) float v2f;
typedef __attribute__((ext_vector_type(4))) float v4f;
typedef __attribute__((ext_vector_type(8))) float v8f;

#define FEAT 64

// ---------------------------------------------------------------------------
// Pass 0: seed h_N with -inf so segment_max + "no-edge -> 0" can be done with
// hardware float-max atomics and a cheap -inf->0 fixup at GEMM load time.
// ---------------------------------------------------------------------------
__global__ void hN_init_kernel(float* __restrict__ hN, int n4) {
    int i = blockIdx.x * blockDim.x + threadIdx.x;
    if (i < n4) {
        float ninf = -__builtin_inff();
        v4f v = {ninf, ninf, ninf, ninf};
        ((v4f*)hN)[i] = v;
    }
}

// ---------------------------------------------------------------------------
// Pass 1: edge-parallel gather + atomic float max (global_atomic_max_num_f32).
// 16 threads per edge, each handling a float4 chunk of the 64-wide feature row.
// This is the bandwidth/atomic-bound part (~307MB gathered reads, 77M atomics).
// ---------------------------------------------------------------------------
__global__ void edge_max_kernel(const float* __restrict__ h,
                                const int* __restrict__ src,
                                const int* __restrict__ dst,
                                float* __restrict__ hN, int nE) {
    long gid = (long)blockIdx.x * blockDim.x + threadIdx.x;
    int e = (int)(gid >> 4);
    int c = (int)(gid & 15);
    if (e >= nE) return;
    int s = src[e];
    int d = dst[e];
    v4f m = ((const v4f*)(h + (long)s * FEAT))[c];
    float* base = hN + (long)d * FEAT + c * 4;
    __hip_atomic_fetch_max(base + 0, m.x, __ATOMIC_RELAXED, __HIP_MEMORY_SCOPE_AGENT);
    __hip_atomic_fetch_max(base + 1, m.y, __ATOMIC_RELAXED, __HIP_MEMORY_SCOPE_AGENT);
    __hip_atomic_fetch_max(base + 2, m.z, __ATOMIC_RELAXED, __HIP_MEMORY_SCOPE_AGENT);
    __hip_atomic_fetch_max(base + 3, m.w, __ATOMIC_RELAXED, __HIP_MEMORY_SCOPE_AGENT);
}

// ---------------------------------------------------------------------------
// Pass 2: out = fix(h_N) @ W^T + b via V_WMMA_F32_16X16X4_F32.
// Block = 256 threads = 8 waves. W (16KB) + b staged in LDS once per block.
// Each wave computes one 16x16 output tile: wave[2..0] -> (mtile offset, jtile).
// A fragment (16x4 f32, 2 VGPR): lane<16 holds K=kb+0,kb+1; lane>=16 K=kb+2,kb+3.
// B fragment symmetric striping with N = lane%16, B[k][j] = W[j][k].
// ---------------------------------------------------------------------------
__global__ void gemm_wmma_kernel(const float* __restrict__ hN,
                                 const float* __restrict__ W,
                                 const float* __restrict__ b,
                                 float* __restrict__ out, int nNodes) {
    __shared__ float sW[FEAT * FEAT];
    __shared__ float sB[FEAT];

    for (int i = threadIdx.x; i < (FEAT * FEAT) / 4; i += blockDim.x)
        ((v4f*)sW)[i] = ((const v4f*)W)[i];
    if (threadIdx.x < FEAT) sB[threadIdx.x] = b[threadIdx.x];
    __syncthreads();

    const int wave = threadIdx.x >> 5;
    const int lane = threadIdx.x & 31;
    const int half = lane >> 4;   // 0: K pair {0,1}, 1: K pair {2,3}
    const int lr   = lane & 15;

    const int mtile = blockIdx.x * 2 + (wave >> 2);
    const int row0  = mtile * 16;
    const int col0  = (wave & 3) * 16;
    if (row0 >= nNodes) return;

    const float bj = sB[col0 + lr];
    const float ninf = -__builtin_inff();

    const float* arow = hN + (long)(row0 + lr) * FEAT + 2 * half; // M = row0+lr
    const float* brow = sW + (col0 + lr) * FEAT + 2 * half;       // N = col0+lr

    v8f acc = {};
#pragma unroll
    for (int kk = 0; kk < 16; ++kk) {
        v2f a = *(const v2f*)(arow + kk * 4);
        // no-incoming-edge rows stayed at -inf -> reference zero-fills them
        a.x = (a.x == ninf) ? 0.0f : a.x;
        a.y = (a.y == ninf) ? 0.0f : a.y;
        v2f bm = *(const v2f*)(brow + kk * 4);
        acc = __builtin_amdgcn_wmma_f32_16x16x4_f32(
            /*neg_a=*/false, a, /*neg_b=*/false, bm,
            /*c_mod=*/(short)0, acc, /*reuse_a=*/false, /*reuse_b=*/false);
    }

    // D layout: VGPR r -> M = r + 8*half, N = lr (within the 16x16 tile)
#pragma unroll
    for (int r = 0; r < 8; ++r) {
        out[(long)(row0 + r + 8 * half) * FEAT + col0 + lr] = acc[r] + bj;
    }
}

// ---------------------------------------------------------------------------
// Pass 3: out[idx] = h[idx]  (row overwrite; duplicate idx writes identical data)
// ---------------------------------------------------------------------------
__global__ void scatter_copy_kernel(const float* __restrict__ h,
                                    const int* __restrict__ idx,
                                    float* __restrict__ out, int nIdx) {
    long gid = (long)blockIdx.x * blockDim.x + threadIdx.x;
    int i = (int)(gid >> 4);
    int c = (int)(gid & 15);
    if (i >= nIdx) return;
    int node = idx[i];
    ((v4f*)(out + (long)node * FEAT))[c] = ((const v4f*)(h + (long)node * FEAT))[c];
}

extern "C" void kernel_launch(void* const* d_in, const int* in_sizes, int n_in,
                              void* d_out, int out_size, void* d_ws, size_t ws_size,
                              hipStream_t stream) {
    const float* h   = (const float*)d_in[0];
    const float* W   = (const float*)d_in[1];
    const float* b   = (const float*)d_in[2];
    const int*   src = (const int*)d_in[3];
    const int*   dst = (const int*)d_in[4];
    const int*   idx = (const int*)d_in[5];
    float* out = (float*)d_out;

    const int nNodes = in_sizes[0] / FEAT;
    const int nE     = in_sizes[3];
    const int nIdx   = in_sizes[5];

    float* hN = (float*)d_ws; // nNodes * FEAT floats of scratch

    // Pass 0: h_N = -inf
    {
        int n4 = (nNodes * FEAT) / 4;
        int blocks = (n4 + 255) / 256;
        hipLaunchKernelGGL(hN_init_kernel, dim3(blocks), dim3(256), 0, stream, hN, n4);
    }
    // Pass 1: segment max via float atomics
    {
        long threads = (long)nE * 16;
        int blocks = (int)((threads + 255) / 256);
        hipLaunchKernelGGL(edge_max_kernel, dim3(blocks), dim3(256), 0, stream,
                           h, src, dst, hN, nE);
    }
    // Pass 2: WMMA GEMM + bias (2 node-tiles x 4 j-tiles per 256-thread block)
    {
        int nTiles = (nNodes + 15) / 16;
        int blocks = (nTiles + 1) / 2;
        hipLaunchKernelGGL(gemm_wmma_kernel, dim3(blocks), dim3(256), 0, stream,
                           hN, W, b, out, nNodes);
    }
    // Pass 3: indexed row overwrite
    {
        long threads = (long)nIdx * 16;
        int blocks = (int)((threads + 255) / 256);
        hipLaunchKernelGGL(scatter_copy_kernel, dim3(blocks), dim3(256), 0, stream,
                           h, idx, out, nIdx);
    }
}